// NCC_82222853914696
// MI455X (gfx1250) — compile-verified
//
#include <hip/hip_runtime.h>

typedef __attribute__((ext_vector_type(2))) float v2f;
typedef __attribute__((ext_vector_type(8))) float v8f;

#define BATCHES 16
#define VOL     (64 * 128 * 128)   /* 1048576 elements per batch volume */
#define GROUPS  (VOL / 4)          /* 262144 float4 groups per batch    */
#define BPB     128                /* blocks per batch                  */
#define THREADS 256                /* 8 wave32 per block                */
#define WAVES   (THREADS / 32)

// Stage 1: per-block partial sums of {I, J, I*I, J*J, I*J, maskI*maskJ}.
// Hierarchy: fp32 thread accumulators -> wave32 shuffle reduce ->
// LDS (8 waves x 6 sums) -> one V_WMMA_F32_16X16X4_F32 with B=ones to fold
// the 8x6 partial matrix into 6 block sums (row sums of A are broadcast to
// every column of D, so D-vgpr q at lane 0 = block sum of accumulator q).
__global__ __launch_bounds__(THREADS)
void ncc_partial_kernel(const float* __restrict__ I,
                        const float* __restrict__ J,
                        const int*  __restrict__ CI,
                        const int*  __restrict__ CJ,
                        float* __restrict__ part)
{
    const int batch = blockIdx.y;
    const size_t base = (size_t)batch * (size_t)VOL;
    const float4* I4  = reinterpret_cast<const float4*>(I  + base);
    const float4* J4  = reinterpret_cast<const float4*>(J  + base);
    const int4*   CI4 = reinterpret_cast<const int4*>(CI + base);
    const int4*   CJ4 = reinterpret_cast<const int4*>(CJ + base);

    float sI = 0.f, sJ = 0.f, sI2 = 0.f, sJ2 = 0.f, sIJ = 0.f, sM = 0.f;

    for (int g = blockIdx.x * THREADS + (int)threadIdx.x; g < GROUPS;
         g += BPB * THREADS) {
        float4 iv = I4[g];
        float4 jv = J4[g];
        int4  civ = CI4[g];
        int4  cjv = CJ4[g];
        float fi[4] = {iv.x, iv.y, iv.z, iv.w};
        float fj[4] = {jv.x, jv.y, jv.z, jv.w};
        int   ci[4] = {civ.x, civ.y, civ.z, civ.w};
        int   cj[4] = {cjv.x, cjv.y, cjv.z, cjv.w};
#pragma unroll
        for (int k = 0; k < 4; ++k) {
            float a = fi[k], b = fj[k];
            sI += a;
            sJ += b;
            sI2 = fmaf(a, a, sI2);
            sJ2 = fmaf(b, b, sJ2);
            sIJ = fmaf(a, b, sIJ);
            float mi = ((a < 0.247f) && (ci[k] > 0)) ? 0.f : 1.f;
            float mj = ((b < 0.247f) && (cj[k] > 0)) ? 0.f : 1.f;
            sM += mi * mj;
        }
    }

    // wave32 reduction (warpSize == 32 on gfx1250)
#pragma unroll
    for (int off = 16; off > 0; off >>= 1) {
        sI  += __shfl_down(sI,  (unsigned)off, 32);
        sJ  += __shfl_down(sJ,  (unsigned)off, 32);
        sI2 += __shfl_down(sI2, (unsigned)off, 32);
        sJ2 += __shfl_down(sJ2, (unsigned)off, 32);
        sIJ += __shfl_down(sIJ, (unsigned)off, 32);
        sM  += __shfl_down(sM,  (unsigned)off, 32);
    }

    __shared__ float wp[WAVES * 6];
    const int lane = threadIdx.x & 31;
    const int wave = threadIdx.x >> 5;
    if (lane == 0) {
        wp[wave * 6 + 0] = sI;
        wp[wave * 6 + 1] = sJ;
        wp[wave * 6 + 2] = sI2;
        wp[wave * 6 + 3] = sJ2;
        wp[wave * 6 + 4] = sIJ;
        wp[wave * 6 + 5] = sM;
    }
    __syncthreads();

    if (wave == 0) {
        // 16x4 fp32 A layout (ISA 7.12.2): lane L in [0,16) supplies
        // A[L][0]/A[L][1] in vgpr0/vgpr1; lane L in [16,32) supplies
        // A[L-16][2]/A[L-16][3]. Row q = accumulator q; each K slot takes a
        // pre-folded pair of wave partials (8 waves -> 4 K slots).
        int q = lane & 15;
        float a0 = 0.f, a1 = 0.f;
        if (q < 6) {
            if (lane < 16) {
                a0 = wp[0 * 6 + q] + wp[1 * 6 + q];
                a1 = wp[2 * 6 + q] + wp[3 * 6 + q];
            } else {
                a0 = wp[4 * 6 + q] + wp[5 * 6 + q];
                a1 = wp[6 * 6 + q] + wp[7 * 6 + q];
            }
        }
        v2f A;  A[0] = a0;  A[1] = a1;
        v2f Bv; Bv[0] = 1.f; Bv[1] = 1.f;   // B = ones => D[m][n] = sum_k A[m][k]
        v8f C = {};
        C = __builtin_amdgcn_wmma_f32_16x16x4_f32(
            /*neg_a=*/false, A, /*neg_b=*/false, Bv,
            /*c_mod=*/(short)0, C, /*reuse_a=*/false, /*reuse_b=*/false);
        if (lane == 0) {
            // D layout: vgpr r, lanes 0-15 hold M=r, N=lane. Lane 0, C[q] is
            // the block sum of accumulator q (q = 0..5 < 8).
            size_t o = ((size_t)batch * BPB + blockIdx.x) * 6;
            part[o + 0] = C[0];
            part[o + 1] = C[1];
            part[o + 2] = C[2];
            part[o + 3] = C[3];
            part[o + 4] = C[4];
            part[o + 5] = C[5];
        }
    }
}

// Stage 2: deterministic fp64 combine of the 16 x 128 x 6 block partials,
// per-batch cc computation, final mean. One block; trivial cost.
__global__ __launch_bounds__(128)
void ncc_final_kernel(const float* __restrict__ part, float* __restrict__ out)
{
    __shared__ double sums[BATCHES][6];
    __shared__ double terms[BATCHES];
    const int t = threadIdx.x;

    if (t < BATCHES * 6) {
        const int b = t / 6, q = t % 6;
        double s = 0.0;
        for (int blk = 0; blk < BPB; ++blk)
            s += (double)part[((size_t)b * BPB + blk) * 6 + q];
        sums[b][q] = s;
    }
    __syncthreads();

    if (t < BATCHES) {
        const double W  = (double)VOL;
        const double Is = sums[t][0], Js = sums[t][1];
        const double I2 = sums[t][2], J2 = sums[t][3];
        const double IJ = sums[t][4], M  = sums[t][5];
        const double uI = Is / W, uJ = Js / W;
        const double cross = IJ - uJ * Is - uI * Js + uI * uJ * W;
        const double Iv = I2 - 2.0 * uI * Is + uI * uI * W;
        const double Jv = J2 - 2.0 * uJ * Js + uJ * uJ * W;
        const double cc = (cross * cross) / (Iv * Jv + 1e-5);
        terms[t] = cc * M;
    }
    __syncthreads();

    if (t == 0) {
        double tot = 0.0;
        for (int b = 0; b < BATCHES; ++b) tot += terms[b];
        out[0] = (float)(-tot / ((double)BATCHES * (double)VOL));
    }
}

extern "C" void kernel_launch(void* const* d_in, const int* in_sizes, int n_in,
                              void* d_out, int out_size, void* d_ws, size_t ws_size,
                              hipStream_t stream)
{
    (void)in_sizes; (void)n_in; (void)out_size; (void)ws_size;
    const float* I  = (const float*)d_in[0];
    const float* J  = (const float*)d_in[1];
    const int*   CI = (const int*)d_in[2];
    const int*   CJ = (const int*)d_in[3];
    float* part = (float*)d_ws;              // 16*128*6 floats = 48 KB scratch
    float* out  = (float*)d_out;

    dim3 grid(BPB, BATCHES);
    ncc_partial_kernel<<<grid, THREADS, 0, stream>>>(I, J, CI, CJ, part);
    ncc_final_kernel<<<1, 128, 0, stream>>>(part, out);
}